// Router_15126874817025
// MI455X (gfx1250) — compile-verified
//
#include <hip/hip_runtime.h>
#include <hip/hip_bf16.h>
#include <math.h>

// MoE router for MI455X (gfx1250, wave32).
// GEMM via V_WMMA_F32_16X16X4_F32 (fp32-exact),
// per-row LayerNorm / temp-softmax / top-2 done with 16-lane xor shuffles on
// the WMMA C/D register layout, aux losses via deterministic 2-phase reduce.

typedef float v2f __attribute__((ext_vector_type(2)));
typedef float v8f __attribute__((ext_vector_type(8)));

#define D_DIM 2048
#define E_DIM 64
#define B_DIM 4
#define S_LEN 4096
#define N_TOK (B_DIM * S_LEN)          // 16384
#define ROWS_PER_WAVE 16
#define WAVES_PER_BLOCK 4
#define NUM_WAVES (N_TOK / ROWS_PER_WAVE)   // 1024
#define WAVES_PER_B (S_LEN / ROWS_PER_WAVE) // 256

__global__ __launch_bounds__(WAVES_PER_BLOCK * 32)
void router_main(const float* __restrict__ x,
                 const float* __restrict__ Wm,
                 const float* __restrict__ gamma,
                 const float* __restrict__ beta,
                 const float* __restrict__ temperature,
                 float* __restrict__ out_rw,
                 float* __restrict__ out_disp,
                 float* __restrict__ ws_z,
                 float* __restrict__ ws_lb)
{
    const int lane = threadIdx.x & 31;
    const int widb = threadIdx.x >> 5;
    const int wave = blockIdx.x * WAVES_PER_BLOCK + widb;
    const int row0 = wave * ROWS_PER_WAVE;
    const int nloc = lane & 15;   // column / row-in-tile selector
    const int hl   = lane >> 4;   // half-wave: selects K pair (A/B), row+8 (C/D)

    // A: lane holds x[row0 + nloc][k + 2*hl + {0,1}] per K-step of 4.
    const float* aptr = x + (size_t)(row0 + nloc) * D_DIM + 2 * hl;
    // B (= W^T): lane holds W[n][k + 2*hl + {0,1}], n = t*16 + nloc.
    const float* bptr = Wm + (size_t)nloc * D_DIM + 2 * hl;

    v8f acc[4];
    acc[0] = (v8f){}; acc[1] = (v8f){}; acc[2] = (v8f){}; acc[3] = (v8f){};

    #pragma unroll 4
    for (int k = 0; k < D_DIM; k += 4) {
        v2f a  = *(const v2f*)(aptr + k);
        v2f b0 = *(const v2f*)(bptr + k);
        v2f b1 = *(const v2f*)(bptr + 16 * D_DIM + k);
        v2f b2 = *(const v2f*)(bptr + 32 * D_DIM + k);
        v2f b3 = *(const v2f*)(bptr + 48 * D_DIM + k);
        acc[0] = __builtin_amdgcn_wmma_f32_16x16x4_f32(false, a, false, b0,
                                                       (short)0, acc[0], false, false);
        acc[1] = __builtin_amdgcn_wmma_f32_16x16x4_f32(false, a, false, b1,
                                                       (short)0, acc[1], false, false);
        acc[2] = __builtin_amdgcn_wmma_f32_16x16x4_f32(false, a, false, b2,
                                                       (short)0, acc[2], false, false);
        acc[3] = __builtin_amdgcn_wmma_f32_16x16x4_f32(false, a, false, b3,
                                                       (short)0, acc[3], false, false);
    }

    // Per-lane LN scale/shift for the 4 expert columns this lane owns.
    float g[4], be[4];
    #pragma unroll
    for (int t = 0; t < 4; ++t) {
        g[t]  = gamma[t * 16 + nloc];
        be[t] = beta [t * 16 + nloc];
    }
    const float tinv = 1.0f / (temperature[0] + 1e-6f);

    float zacc = 0.0f;
    float lb[4] = {0.f, 0.f, 0.f, 0.f};

    // C/D layout: reg r, lanes 0-15 -> row r, lanes 16-31 -> row r+8; col = nloc.
    #pragma unroll
    for (int r = 0; r < 8; ++r) {
        float v[4];
        float sum = 0.f, sumsq = 0.f;
        #pragma unroll
        for (int t = 0; t < 4; ++t) {
            v[t] = acc[t][r];
            sum += v[t];
            sumsq += v[t] * v[t];
        }
        // Reduce across the 16 lanes that share this row (masks 1,2,4,8 only).
        #pragma unroll
        for (int m = 1; m <= 8; m <<= 1) {
            sum   += __shfl_xor(sum,   m);
            sumsq += __shfl_xor(sumsq, m);
        }
        float mean = sum * (1.0f / 64.0f);
        float var  = fmaxf(sumsq * (1.0f / 64.0f) - mean * mean, 0.0f);
        float rstd = rsqrtf(var + 1e-5f);

        float ln[4], sm[4];
        float rmax = -1e30f;
        #pragma unroll
        for (int t = 0; t < 4; ++t) {
            ln[t] = (v[t] - mean) * rstd * g[t] + be[t];
            zacc += ln[t] * ln[t];
            sm[t] = ln[t] * tinv;
            rmax = fmaxf(rmax, sm[t]);
        }
        #pragma unroll
        for (int m = 1; m <= 8; m <<= 1)
            rmax = fmaxf(rmax, __shfl_xor(rmax, m));

        float e[4], es = 0.f;
        #pragma unroll
        for (int t = 0; t < 4; ++t) {
            e[t] = __expf(sm[t] - rmax);
            es += e[t];
        }
        #pragma unroll
        for (int m = 1; m <= 8; m <<= 1)
            es += __shfl_xor(es, m);
        float inv = 1.0f / es;
        float w[4];
        #pragma unroll
        for (int t = 0; t < 4; ++t) w[t] = e[t] * inv;

        // Per-lane top-2 (value desc, index asc on ties), then merge across lanes.
        float w1 = -1e30f, w2 = -1e30f;
        int   i1 = 0,      i2 = 0;
        #pragma unroll
        for (int t = 0; t < 4; ++t) {
            int idx = t * 16 + nloc;
            if (w[t] > w1) { w2 = w1; i2 = i1; w1 = w[t]; i1 = idx; }
            else if (w[t] > w2) { w2 = w[t]; i2 = idx; }
        }
        #pragma unroll
        for (int m = 1; m <= 8; m <<= 1) {
            float ow1 = __shfl_xor(w1, m); int oi1 = __shfl_xor(i1, m);
            float ow2 = __shfl_xor(w2, m); int oi2 = __shfl_xor(i2, m);
            bool ofirst = (ow1 > w1) || (ow1 == w1 && oi1 < i1);
            float n1, n2; int j1, j2;
            if (ofirst) {
                n1 = ow1; j1 = oi1;
                if ((w1 > ow2) || (w1 == ow2 && i1 < oi2)) { n2 = w1;  j2 = i1;  }
                else                                       { n2 = ow2; j2 = oi2; }
            } else {
                n1 = w1; j1 = i1;
                if ((ow1 > w2) || (ow1 == w2 && oi1 < i2)) { n2 = ow1; j2 = oi1; }
                else                                       { n2 = w2;  j2 = i2;  }
            }
            w1 = n1; i1 = j1; w2 = n2; i2 = j2;
        }

        const int R = row0 + r + 8 * hl;
        const size_t base = (size_t)R * E_DIM + nloc;
        #pragma unroll
        for (int t = 0; t < 4; ++t) {
            int idx = t * 16 + nloc;
            float dv = (idx == i1 || idx == i2) ? w[t] : 0.0f;
            out_rw [base + t * 16] = w[t];
            out_disp[base + t * 16] = dv;
            lb[t] += dv;
        }
    }

    // Load-balance partials: add the other half's 8 rows, write once per expert.
    #pragma unroll
    for (int t = 0; t < 4; ++t)
        lb[t] += __shfl_xor(lb[t], 16);
    if (hl == 0) {
        #pragma unroll
        for (int t = 0; t < 4; ++t)
            ws_lb[(size_t)wave * E_DIM + t * 16 + nloc] = lb[t];
    }

    // z-loss partial: full-wave reduction.
    #pragma unroll
    for (int m = 1; m <= 16; m <<= 1)
        zacc += __shfl_xor(zacc, m);
    if (lane == 0) ws_z[wave] = zacc;
}

__global__ __launch_bounds__(256)
void router_finalize(const float* __restrict__ ws_z,
                     const float* __restrict__ ws_lb,
                     float* __restrict__ out_loss)
{
    __shared__ float s_load[256];
    __shared__ float s_z[256];
    const int tid = threadIdx.x;

    float z = 0.f;
    #pragma unroll
    for (int i = 0; i < 4; ++i) z += ws_z[tid * 4 + i];
    s_z[tid] = z;

    const int b = tid >> 6;
    const int e = tid & 63;
    float accv = 0.f;
    for (int w = 0; w < WAVES_PER_B; ++w)
        accv += ws_lb[(size_t)(b * WAVES_PER_B + w) * E_DIM + e];
    s_load[tid] = accv / (float)S_LEN;   // expert_load[b][e]
    __syncthreads();

    if (tid == 0) {
        float zsum = 0.f;
        for (int i = 0; i < 256; ++i) zsum += s_z[i];
        float zloss = zsum / (float)((size_t)N_TOK * E_DIM);

        float lsum = 0.f;
        for (int i = 0; i < 256; ++i) lsum += s_load[i];
        float mean = lsum / 256.0f;
        float sq = 0.f;
        for (int i = 0; i < 256; ++i) {
            float d = s_load[i] - mean;
            sq += d * d;
        }
        float stdv = sqrtf(sq / 255.0f);         // ddof=1
        float lb_loss = stdv / mean * 10.0f;
        out_loss[0] = 0.001f * zloss + 0.1f * lb_loss;
    }
}

extern "C" void kernel_launch(void* const* d_in, const int* in_sizes, int n_in,
                              void* d_out, int out_size, void* d_ws, size_t ws_size,
                              hipStream_t stream)
{
    const float* x     = (const float*)d_in[0];
    const float* Wm    = (const float*)d_in[1];
    const float* gamma = (const float*)d_in[2];
    const float* beta  = (const float*)d_in[3];
    const float* temp  = (const float*)d_in[4];

    float* out      = (float*)d_out;
    float* out_rw   = out;                                 // [N, E]
    float* out_disp = out + (size_t)N_TOK * E_DIM;         // [B, S, E]
    float* out_loss = out + 2 * (size_t)N_TOK * E_DIM;     // scalar

    float* ws_z  = (float*)d_ws;          // NUM_WAVES floats
    float* ws_lb = (float*)d_ws + 1024;   // NUM_WAVES * E floats

    const int blocks = N_TOK / (ROWS_PER_WAVE * WAVES_PER_BLOCK);  // 256
    router_main<<<blocks, WAVES_PER_BLOCK * 32, 0, stream>>>(
        x, Wm, gamma, beta, temp, out_rw, out_disp, ws_z, ws_lb);
    router_finalize<<<1, 256, 0, stream>>>(ws_z, ws_lb, out_loss);
}